// GNN_61555471286611
// MI455X (gfx1250) — compile-verified
//
#include <hip/hip_runtime.h>
#include <hip/hip_bf16.h>

// ---------------------------------------------------------------------------
// 2-layer GAT for MI455X (gfx1250, wave32).
//   - h = x@W done with V_WMMA_F32_16X16X4_F32 (full-fp32 matrix core path)
//   - layer-2 input ReLU folded into the WMMA A-fragment load at COMPILE time
//     (template) so layer-1's inner loop is pure load->wmma
//   - edge softmax via ordered-uint float atomicMax + float atomicAdd
//   - aggregation: one wave32 per edge, lane c handles channels c and c+32
// Working set (h, out, edge arrays ~90 MB) is L2-resident (192 MB L2), so the
// random gather/scatter over 1.6M edges stays on-chip; HBM sees only the
// initial streaming pass.
// ---------------------------------------------------------------------------

#define GAT_N 100000
#define GAT_E 1600000
#define GAT_C 64
#define NEG_SLOPE 0.2f

typedef __attribute__((ext_vector_type(2))) float v2f;
typedef __attribute__((ext_vector_type(8))) float v8f;

// ---- ordered-uint encoding for float atomic max ---------------------------
__device__ __forceinline__ unsigned enc_f32(float f) {
    unsigned u = __float_as_uint(f);
    return u ^ ((((int)u) >> 31) | 0x80000000u);   // monotonic float->uint
}
__device__ __forceinline__ float dec_f32(unsigned k) {
    unsigned u = (k & 0x80000000u) ? (k ^ 0x80000000u) : ~k;
    return __uint_as_float(u);
}

// ---- scalar ce = dot(We[0,:], a_e) ----------------------------------------
__global__ void gat_ce_kernel(const float* __restrict__ We,
                              const float* __restrict__ a_e,
                              float* __restrict__ ce) {
    if (threadIdx.x == 0) {
        float s = 0.f;
        #pragma unroll
        for (int c = 0; c < GAT_C; ++c) s += We[c] * a_e[c];
        ce[0] = s;
    }
}

// ---- H = X @ W via WMMA f32 16x16x4 ---------------------------------------
// Block = 128 threads = 4 waves. Block b covers node rows [16b,16b+16);
// wave w covers output columns [16w,16w+16). K=64 -> 16 WMMA ops per wave.
// RELU_IN is compile-time: layer 1 instantiates the clean no-relu version.
template <bool RELU_IN>
__global__ void __launch_bounds__(128)
gat_gemm_wmma(const float* __restrict__ X, const float* __restrict__ W,
              float* __restrict__ H) {
    const int wave = threadIdx.x >> 5;        // 0..3
    const int lane = threadIdx.x & 31;
    const int half = lane >> 4;               // 0: lanes 0-15, 1: lanes 16-31
    const int l16  = lane & 15;
    const int rowBase = blockIdx.x * 16;      // N == 100000 is 16-aligned
    const int colBase = wave * 16;

    v8f acc = {};
    const int arow = rowBase + l16;
    #pragma unroll
    for (int k = 0; k < GAT_C; k += 4) {
        // A fragment: 16x4 tile of X. lane<16 holds K=k..k+1, lane>=16 K=k+2..k+3
        const int ka = k + half * 2;
        float ax = X[arow * GAT_C + ka];
        float ay = X[arow * GAT_C + ka + 1];
        if (RELU_IN) { ax = fmaxf(ax, 0.f); ay = fmaxf(ay, 0.f); }
        v2f a = {ax, ay};
        // B fragment: 4x16 tile of W (row-major [K][N])
        v2f b = { W[ka * GAT_C + colBase + l16],
                  W[(ka + 1) * GAT_C + colBase + l16] };
        acc = __builtin_amdgcn_wmma_f32_16x16x4_f32(
            /*neg_a=*/false, a, /*neg_b=*/false, b,
            /*c_mod=*/(short)0, acc, /*reuse_a=*/false, /*reuse_b=*/false);
    }
    // C/D layout: VGPR r holds M = r + 8*half, N = l16
    const int mrow = rowBase + half * 8;
    #pragma unroll
    for (int r = 0; r < 8; ++r)
        H[(mrow + r) * GAT_C + colBase + l16] = acc[r];
}

// ---- per-node alpha_src/alpha_dst dots + softmax-state init ----------------
__global__ void gat_node_alpha(const float* __restrict__ H,
                               const float* __restrict__ a_src,
                               const float* __restrict__ a_dst,
                               float* __restrict__ asrc, float* __restrict__ adst,
                               unsigned* __restrict__ mkey, float* __restrict__ ssum) {
    int i = blockIdx.x * blockDim.x + threadIdx.x;
    if (i >= GAT_N) return;
    const float4* hp = (const float4*)(H + (size_t)i * GAT_C);
    const float4* as = (const float4*)a_src;
    const float4* ad = (const float4*)a_dst;
    float s0 = 0.f, s1 = 0.f;
    #pragma unroll
    for (int j = 0; j < GAT_C / 4; ++j) {
        float4 h4 = hp[j], a4 = as[j], d4 = ad[j];
        s0 += h4.x * a4.x + h4.y * a4.y + h4.z * a4.z + h4.w * a4.w;
        s1 += h4.x * d4.x + h4.y * d4.y + h4.z * d4.z + h4.w * d4.w;
    }
    asrc[i] = s0;
    adst[i] = s1;
    mkey[i] = 0u;     // encodes "smaller than any real alpha"
    ssum[i] = 0.f;
}

// ---- edge alpha + leaky relu + segment max --------------------------------
__global__ void gat_edge_alpha_max(const long long* __restrict__ src,
                                   const long long* __restrict__ dst,
                                   const float* __restrict__ eattr,
                                   const float* __restrict__ asrc,
                                   const float* __restrict__ adst,
                                   const float* __restrict__ ce,
                                   float* __restrict__ ew,
                                   unsigned* __restrict__ mkey) {
    int e = blockIdx.x * blockDim.x + threadIdx.x;
    if (e >= GAT_E) return;
    long long s = src[e], d = dst[e];
    float a = asrc[s] + adst[d] + eattr[e] * ce[0];
    a = (a > 0.f) ? a : NEG_SLOPE * a;               // leaky relu
    ew[e] = a;
    atomicMax(&mkey[d], enc_f32(a));                 // global_atomic_max_u32
}

// ---- exp(alpha - m[dst]) + segment sum ------------------------------------
__global__ void gat_edge_exp_sum(const long long* __restrict__ dst,
                                 const unsigned* __restrict__ mkey,
                                 float* __restrict__ ew,
                                 float* __restrict__ ssum) {
    int e = blockIdx.x * blockDim.x + threadIdx.x;
    if (e >= GAT_E) return;
    long long d = dst[e];
    float w = __expf(ew[e] - dec_f32(mkey[d]));
    ew[e] = w;
    atomicAdd(&ssum[d], w);                          // global_atomic_add_f32
}

// ---- out init: broadcast bias ---------------------------------------------
__global__ void gat_init_out(float* __restrict__ out, const float* __restrict__ b) {
    int i = blockIdx.x * blockDim.x + threadIdx.x;
    if (i < GAT_N * GAT_C) out[i] = b[i & (GAT_C - 1)];
}

// ---- aggregation: out[dst] += (w/s) * h[src] ------------------------------
// One wave32 per edge; lane c does channels c and c+32.
__global__ void gat_aggregate(const long long* __restrict__ src,
                              const long long* __restrict__ dst,
                              const float* __restrict__ ew,
                              const float* __restrict__ ssum,
                              const float* __restrict__ H,
                              float* __restrict__ out) {
    long long tid = (long long)blockIdx.x * blockDim.x + threadIdx.x;
    int e = (int)(tid >> 5);
    int lane = (int)(tid & 31);
    if (e >= GAT_E) return;
    long long s = src[e], d = dst[e];
    float coeff = ew[e] / (ssum[d] + 1e-16f);
    const float* hs = H + (size_t)s * GAT_C;
    float* od = out + (size_t)d * GAT_C;
    atomicAdd(&od[lane],      coeff * hs[lane]);
    atomicAdd(&od[lane + 32], coeff * hs[lane + 32]);
}

// ---------------------------------------------------------------------------
// Workspace layout
// ---------------------------------------------------------------------------
struct GatWs {
    float*    h;      // [N*C]
    float*    l1out;  // [N*C]
    float*    asrc;   // [N]
    float*    adst;   // [N]
    unsigned* mkey;   // [N]
    float*    ssum;   // [N]
    float*    ew;     // [E]
    float*    ce;     // [1]
};

static inline size_t align256(size_t x) { return (x + 255) & ~(size_t)255; }

template <bool RELU_IN>
static void run_gat_layer(const float* xin,
                          const long long* src, const long long* dst,
                          const float* eattr,
                          const float* W, const float* a_src, const float* a_dst,
                          const float* We, const float* a_e, const float* bias,
                          float* out, const GatWs& ws, hipStream_t stream) {
    gat_ce_kernel<<<1, 64, 0, stream>>>(We, a_e, ws.ce);
    gat_gemm_wmma<RELU_IN><<<GAT_N / 16, 128, 0, stream>>>(xin, W, ws.h);
    gat_node_alpha<<<(GAT_N + 255) / 256, 256, 0, stream>>>(
        ws.h, a_src, a_dst, ws.asrc, ws.adst, ws.mkey, ws.ssum);
    gat_edge_alpha_max<<<(GAT_E + 255) / 256, 256, 0, stream>>>(
        src, dst, eattr, ws.asrc, ws.adst, ws.ce, ws.ew, ws.mkey);
    gat_edge_exp_sum<<<(GAT_E + 255) / 256, 256, 0, stream>>>(
        dst, ws.mkey, ws.ew, ws.ssum);
    gat_init_out<<<(GAT_N * GAT_C + 255) / 256, 256, 0, stream>>>(out, bias);
    long long aggThreads = (long long)GAT_E * 32;
    gat_aggregate<<<(unsigned)((aggThreads + 255) / 256), 256, 0, stream>>>(
        src, dst, ws.ew, ws.ssum, ws.h, out);
}

extern "C" void kernel_launch(void* const* d_in, const int* in_sizes, int n_in,
                              void* d_out, int out_size, void* d_ws, size_t ws_size,
                              hipStream_t stream) {
    (void)in_sizes; (void)n_in; (void)out_size; (void)ws_size;

    const float*     x     = (const float*)d_in[0];
    const long long* eidx  = (const long long*)d_in[1];   // int64 [2,E]
    const float*     eattr = (const float*)d_in[2];       // [E,1]
    const float* W1 = (const float*)d_in[3];
    const float* as1 = (const float*)d_in[4];
    const float* ad1 = (const float*)d_in[5];
    const float* We1 = (const float*)d_in[6];
    const float* ae1 = (const float*)d_in[7];
    const float* b1  = (const float*)d_in[8];
    const float* W2 = (const float*)d_in[9];
    const float* as2 = (const float*)d_in[10];
    const float* ad2 = (const float*)d_in[11];
    const float* We2 = (const float*)d_in[12];
    const float* ae2 = (const float*)d_in[13];
    const float* b2  = (const float*)d_in[14];

    const long long* src = eidx;
    const long long* dst = eidx + GAT_E;

    // carve workspace
    char* p = (char*)d_ws;
    GatWs ws;
    ws.h     = (float*)p;    p += align256((size_t)GAT_N * GAT_C * 4);
    ws.l1out = (float*)p;    p += align256((size_t)GAT_N * GAT_C * 4);
    ws.asrc  = (float*)p;    p += align256((size_t)GAT_N * 4);
    ws.adst  = (float*)p;    p += align256((size_t)GAT_N * 4);
    ws.mkey  = (unsigned*)p; p += align256((size_t)GAT_N * 4);
    ws.ssum  = (float*)p;    p += align256((size_t)GAT_N * 4);
    ws.ew    = (float*)p;    p += align256((size_t)GAT_E * 4);
    ws.ce    = (float*)p;    p += align256(4);

    // layer 1: x -> l1out  (no input relu)
    run_gat_layer<false>(x, src, dst, eattr,
                         W1, as1, ad1, We1, ae1, b1, ws.l1out, ws, stream);
    // layer 2: relu(l1out) -> d_out   (relu folded into WMMA A-load)
    run_gat_layer<true>(ws.l1out, src, dst, eattr,
                        W2, as2, ad2, We2, ae2, b2, (float*)d_out, ws, stream);
}